// NCA_70033736729268
// MI455X (gfx1250) — compile-verified
//
#include <hip/hip_runtime.h>

typedef __attribute__((ext_vector_type(2))) float v2f;
typedef __attribute__((ext_vector_type(8))) float v8f;

#define Hdim 256
#define Wdim 256
#define Cdim 16
#define HIDN 128

// ---------------- Pass 1: perception + 2-layer MLP via f32 WMMA ----------------
// Grid: (W/16, H/8, B). Block: 256 threads = 8 waves; wave w handles image row
// (row0+w), 16 consecutive columns. Pixels are the WMMA N dimension.
__global__ __launch_bounds__(256) void nca_update_kernel(
    const float* __restrict__ x, const float* __restrict__ urand,
    const float* __restrict__ w1, const float* __restrict__ b1,
    const float* __restrict__ w2, float* __restrict__ out,
    float* __restrict__ ch3_plane)
{
    __shared__ float w1s[HIDN * 48];      // 6144 f
    __shared__ float w2s[Cdim * HIDN];    // 2048 f
    __shared__ float b1s[HIDN];           //  128 f
    __shared__ float xs[Cdim * 10 * 18];  // 2880 f : x tile + 1-halo
    __shared__ float hs[8 * 16 * 17];     // 2176 f : per-wave h scratch (pad 17)

    const int tid  = threadIdx.x;
    const int b    = blockIdx.z;
    const int row0 = blockIdx.y * 8;
    const int col0 = blockIdx.x * 16;

    // ---- cooperative staging ----
    for (int i = tid; i < HIDN * 48; i += 256)  w1s[i] = w1[i];
    for (int i = tid; i < Cdim * HIDN; i += 256) w2s[i] = w2[i];
    if (tid < HIDN) b1s[tid] = b1[tid];
    for (int i = tid; i < Cdim * 10 * 18; i += 256) {
        int c = i / 180, rem = i % 180;
        int rr = rem / 18, cc = rem % 18;
        int gh = row0 - 1 + rr, gw = col0 - 1 + cc;
        float v = 0.0f;  // conv zero padding at image border
        if (gh >= 0 && gh < Hdim && gw >= 0 && gw < Wdim)
            v = x[(((size_t)b * Cdim + c) * Hdim + gh) * Wdim + gw];
        xs[i] = v;
    }
    __syncthreads();

    const int wave = tid >> 5;
    const int lane = tid & 31;
    const int p    = lane & 15;   // pixel within tile (N index)
    const int half = lane >> 4;   // 0: lanes 0-15, 1: lanes 16-31

    // ---- perception, built directly in WMMA B-layout ----
    // by[kk] holds K rows {4kk+2*half, 4kk+2*half+1} for column p.
    v2f by[12];
    #pragma unroll
    for (int kk = 0; kk < 12; ++kk) {
        #pragma unroll
        for (int q = 0; q < 2; ++q) {
            int krow = 4 * kk + 2 * half + q;
            int c = krow / 3, j = krow % 3;
            const float* xc = &xs[(c * 10 + wave + 1) * 18 + (p + 1)];
            float v;
            if (j == 0) {
                v = xc[0];                                   // identity
            } else if (j == 1) {                             // sobel-x / 8
                v = ((xc[-18 + 1] - xc[-18 - 1])
                     + 2.0f * (xc[1] - xc[-1])
                     + (xc[18 + 1] - xc[18 - 1])) * 0.125f;
            } else {                                         // sobel-y / 8
                v = ((xc[18 - 1] - xc[-18 - 1])
                     + 2.0f * (xc[18] - xc[-18])
                     + (xc[18 + 1] - xc[-18 + 1])) * 0.125f;
            }
            by[kk][q] = v;
        }
    }

    // ---- fused layer1 (48->128) + layer2 (128->16) over 8 hid-tiles ----
    v8f dxc = {0.f, 0.f, 0.f, 0.f, 0.f, 0.f, 0.f, 0.f};   // dx^T tile (16 out x 16 pix)
    float* hw = &hs[wave * (16 * 17)];

    for (int t = 0; t < 8; ++t) {
        // init accumulator with bias (row = 16t + r + 8*half)
        v8f hc;
        #pragma unroll
        for (int r = 0; r < 8; ++r) hc[r] = b1s[16 * t + r + 8 * half];

        // layer 1: h_tile(16x16) = W1_tile(16x48) @ y(48x16)
        #pragma unroll
        for (int kk = 0; kk < 12; ++kk) {
            int arow = 16 * t + p;              // M = hid row (lane&15 for both halves)
            int acol = 4 * kk + 2 * half;       // K base per half
            v2f a = *(const v2f*)&w1s[arow * 48 + acol];
            hc = __builtin_amdgcn_wmma_f32_16x16x4_f32(
                    false, a, false, by[kk], (short)0, hc, false, false);
        }
        #pragma unroll
        for (int r = 0; r < 8; ++r) hc[r] = fmaxf(hc[r], 0.0f);  // ReLU

        // C-layout -> LDS -> B-layout round trip (in-wave DS ordering)
        #pragma unroll
        for (int r = 0; r < 8; ++r)
            hw[(r + 8 * half) * 17 + p] = hc[r];
        asm volatile("s_wait_dscnt 0" ::: "memory");

        // layer 2: dx^T += W2_cols(16x16) @ h_tile(16x16), 4 K-steps of 4
        #pragma unroll
        for (int kk = 0; kk < 4; ++kk) {
            int k0 = 4 * kk + 2 * half;
            v2f bh;
            bh[0] = hw[k0 * 17 + p];
            bh[1] = hw[(k0 + 1) * 17 + p];
            int acol = 16 * t + 4 * kk + 2 * half;
            v2f a = *(const v2f*)&w2s[p * HIDN + acol];   // M = out channel
            dxc = __builtin_amdgcn_wmma_f32_16x16x4_f32(
                    false, a, false, bh, (short)0, dxc, false, false);
        }
    }

    // ---- x_new = x + dx * fire ----
    const int gh = row0 + wave;
    const int gw = col0 + p;
    float u    = urand[((size_t)b * Hdim + gh) * Wdim + gw];
    float fire = (u <= 0.5f) ? 1.0f : 0.0f;
    #pragma unroll
    for (int r = 0; r < 8; ++r) {
        int ch = r + 8 * half;
        float xv = xs[(ch * 10 + wave + 1) * 18 + (p + 1)];
        float xn = xv + dxc[r] * fire;
        out[(((size_t)b * Cdim + ch) * Hdim + gh) * Wdim + gw] = xn;
        if (ch == 3)
            ch3_plane[((size_t)b * Hdim + gh) * Wdim + gw] = xn;  // for pass 2 (race-free)
    }
}

// ---------------- Pass 2: alive mask, in-place over d_out ----------------
__global__ __launch_bounds__(256) void nca_alive_kernel(
    const float* __restrict__ x, const float* __restrict__ ch3_plane,
    float* __restrict__ out)
{
    int pix = blockIdx.x * 256 + threadIdx.x;   // b*65536 + hh*256 + ww
    int b   = pix >> 16;
    int rem = pix & 65535;
    int hh  = rem >> 8, ww = rem & 255;

    const float NEG = -3.4e38f;                 // -inf padding for maxpool
    float m_old = NEG, m_new = NEG;
    const float* x3 = &x[(((size_t)b * Cdim) + 3) * (size_t)(Hdim * Wdim)];
    const float* n3 = &ch3_plane[(size_t)b * (Hdim * Wdim)];
    #pragma unroll
    for (int dr = -1; dr <= 1; ++dr) {
        #pragma unroll
        for (int dc = -1; dc <= 1; ++dc) {
            int nh = hh + dr, nw = ww + dc;
            if (nh >= 0 && nh < Hdim && nw >= 0 && nw < Wdim) {
                m_old = fmaxf(m_old, x3[nh * Wdim + nw]);
                m_new = fmaxf(m_new, n3[nh * Wdim + nw]);
            }
        }
    }
    float mask = (m_old > 0.1f && m_new > 0.1f) ? 1.0f : 0.0f;

    size_t base = ((size_t)b * Cdim) * (size_t)(Hdim * Wdim) + (size_t)hh * Wdim + ww;
    #pragma unroll
    for (int c = 0; c < Cdim; ++c)
        out[base + (size_t)c * (Hdim * Wdim)] *= mask;
}

extern "C" void kernel_launch(void* const* d_in, const int* in_sizes, int n_in,
                              void* d_out, int out_size, void* d_ws, size_t ws_size,
                              hipStream_t stream) {
    const float* x     = (const float*)d_in[0];   // (32,16,256,256)
    const float* urand = (const float*)d_in[1];   // (32,1,256,256)
    const float* w1    = (const float*)d_in[2];   // (128,48)
    const float* b1    = (const float*)d_in[3];   // (128,)
    const float* w2    = (const float*)d_in[4];   // (16,128)
    float* out   = (float*)d_out;                 // (32,16,256,256)
    float* plane = (float*)d_ws;                  // 32*256*256 floats = 8 MB

    dim3 grid1(Wdim / 16, Hdim / 8, 32);
    nca_update_kernel<<<grid1, 256, 0, stream>>>(x, urand, w1, b1, w2, out, plane);

    int npix = 32 * Hdim * Wdim;
    nca_alive_kernel<<<npix / 256, 256, 0, stream>>>(x, plane, out);
}